// GNN_multiple_77764677861852
// MI455X (gfx1250) — compile-verified
//
#include <hip/hip_runtime.h>
#include <math.h>

// ---------------- problem constants (match reference) ----------------
#define NN   50000
#define EE   800000
#define BB   256
#define FEAT 128        // IN == HID == 128
#define OUTC 10
#define EPSV 1e-5f

typedef __attribute__((ext_vector_type(2))) float v2f;
typedef __attribute__((ext_vector_type(8))) float v8f;
typedef __attribute__((ext_vector_type(4))) unsigned int u32x4;
typedef __attribute__((ext_vector_type(8))) int i32x8;
typedef __attribute__((ext_vector_type(4))) int i32x4;

// ---------------- utility ----------------
__global__ void zero_f32(float* __restrict__ p, int n) {
    int t = blockIdx.x * blockDim.x + threadIdx.x;
    if (t < n) p[t] = 0.0f;
}

// degree accumulation: one thread per edge
__global__ void deg_count(const int* __restrict__ src, const int* __restrict__ dst,
                          float* __restrict__ degout, float* __restrict__ degin) {
    int e = blockIdx.x * blockDim.x + threadIdx.x;
    if (e < EE) {
        atomicAdd(&degout[src[e]], 1.0f);
        atomicAdd(&degin[dst[e]], 1.0f);
    }
}

// rd = clip(deg,1)^-1/2, in place
__global__ void deg_finalize(float* __restrict__ rdout, float* __restrict__ rdin) {
    int i = blockIdx.x * blockDim.x + threadIdx.x;
    if (i < NN) {
        rdout[i] = rsqrtf(fmaxf(rdout[i], 1.0f));
        rdin[i]  = rsqrtf(fmaxf(rdin[i],  1.0f));
    }
}

// ---------------- WMMA fp32 GEMM: H[N,128] = X[N,128] @ W[128,128] ----------------
// 256 threads = 8 waves per block; W staged into LDS (64 KB) by the Tensor Data
// Mover (one DMA per block, TENSORcnt-tracked); each wave produces a 16-row x
// 128-col slab using v_wmma_f32_16x16x4_f32 (32 k-steps x 8 n-tiles).
__global__ __launch_bounds__(256)
void gemm_wmma_f32(const float* __restrict__ X, const float* __restrict__ Wm,
                   float* __restrict__ H, int nrows) {
    __shared__ float lw[FEAT * FEAT];

#if defined(__gfx1250__) && __has_builtin(__builtin_amdgcn_tensor_load_to_lds)
    if (threadIdx.x == 0) {
        // Tensor DMA descriptor (ISA cdna5 ch.8): 2D 128x128 f32 tensor == tile.
        const unsigned long long ga = (unsigned long long)(uintptr_t)Wm;
        const unsigned ldsoff = (unsigned)(uintptr_t)(&lw[0]);   // LDS addr = low 32 bits
        u32x4 g0;
        g0[0] = 1u;                                   // count=1, is_restore=0, gather off
        g0[1] = ldsoff;                               // lds_addr (bytes)
        g0[2] = (unsigned)(ga & 0xFFFFFFFFull);       // global_addr[31:0]
        g0[3] = (unsigned)((ga >> 32) & 0x01FFFFFFu)  // global_addr[56:32]
                | (2u << 30);                         // type=2 ("image")
        i32x8 g1;
        g1[0] = 2 << 16;                              // workgroup_mask=0, data_size=2 (4B)
        g1[1] = (int)(128u << 16);                    // tensor_dim0 = 128 (low 16 in [63:48])
        g1[2] = (int)(128u << 16);                    // dim0 hi=0, tensor_dim1 = 128 (low 16 in [95:80])
        g1[3] = (int)(128u << 16);                    // dim1 hi=0, tile_dim0 = 128 ([127:112])
        g1[4] = 128;                                  // tile_dim1 = 128 ([143:128]), tile_dim2 = 0
        g1[5] = 128;                                  // tensor_dim0_stride = 128 ([207:160] low 32)
        g1[6] = 0;                                    // stride0 hi, tensor_dim1_stride lo (unused)
        g1[7] = 0;
        i32x4 gz = {0, 0, 0, 0};
#if __clang_major__ >= 23
        i32x8 gz8 = {0, 0, 0, 0, 0, 0, 0, 0};
        __builtin_amdgcn_tensor_load_to_lds(g0, g1, gz, gz, gz8, 0);
#else
        __builtin_amdgcn_tensor_load_to_lds(g0, g1, gz, gz, 0);
#endif
#if __has_builtin(__builtin_amdgcn_s_wait_tensorcnt)
        __builtin_amdgcn_s_wait_tensorcnt(0);
#else
        asm volatile("s_wait_tensorcnt 0x0" ::: "memory");
#endif
    }
#else
    // fallback: cooperative vector load of W (16384 floats = 4096 float4)
    {
        const float4* w4 = (const float4*)Wm;
        float4* l4 = (float4*)lw;
        for (int i = threadIdx.x; i < (FEAT * FEAT / 4); i += 256) l4[i] = w4[i];
    }
#endif
    __syncthreads();

    const int lane = threadIdx.x & 31;
    const int wave = threadIdx.x >> 5;
    const int tile = blockIdx.x * 8 + wave;       // 16-row tile index
    if (tile * 16 >= nrows) return;

    const int m  = lane & 15;                     // row-in-tile (A), col-in-tile (B/D)
    const int hi = lane >> 4;                     // half-wave selects K pair / M+8

    const float2* xr2 = (const float2*)(X + (size_t)(tile * 16 + m) * FEAT);

    v8f acc[8] = {};                              // 8 column tiles of 16
    for (int k = 0; k < FEAT; k += 4) {
        // A fragment: lanes 0-15 -> K=k,k+1 ; lanes 16-31 -> K=k+2,k+3  (ISA 7.12.2)
        float2 af = xr2[(k >> 1) + hi];
        v2f a; a[0] = af.x; a[1] = af.y;
        const float* wrow0 = &lw[(k + 2 * hi) * FEAT];
#pragma unroll
        for (int n = 0; n < 8; ++n) {
            const int col = n * 16 + m;
            // B fragment 4x16: VGPR0 = row K (lanes 0-15) / K+2 (lanes 16-31), VGPR1 = K+1 / K+3
            v2f b; b[0] = wrow0[col]; b[1] = wrow0[FEAT + col];
            acc[n] = __builtin_amdgcn_wmma_f32_16x16x4_f32(
                false, a, false, b, (short)0, acc[n], false, false);
        }
    }
    // D layout: VGPR r -> row r (lanes 0-15) / row r+8 (lanes 16-31), col = lane&15
#pragma unroll
    for (int n = 0; n < 8; ++n) {
#pragma unroll
        for (int r = 0; r < 8; ++r) {
            H[(size_t)(tile * 16 + r + 8 * hi) * FEAT + n * 16 + m] = acc[n][r];
        }
    }
}

// ---------------- edge message + scatter-add ----------------
// one wave per edge; each lane handles a 4-float feature chunk
__global__ __launch_bounds__(256)
void edge_scatter(const float* __restrict__ H, const int* __restrict__ src,
                  const int* __restrict__ dst, const float* __restrict__ rdout,
                  float* __restrict__ AGG) {
    const long t = (long)blockIdx.x * 256 + threadIdx.x;
    const int e = (int)(t >> 5);
    if (e >= EE) return;
    const int c = (int)(t & 31) << 2;             // feature offset (0..124)
    const int s = src[e];
    const int d = dst[e];
    const float sc = rdout[s];
    const float4 h4 = *(const float4*)(H + (size_t)s * FEAT + c);
    float* out = AGG + (size_t)d * FEAT + c;
    atomicAdd(out + 0, h4.x * sc);
    atomicAdd(out + 1, h4.y * sc);
    atomicAdd(out + 2, h4.z * sc);
    atomicAdd(out + 3, h4.w * sc);
}

// h = relu(AGG * rdin + bias), elementwise
__global__ void conv_finalize_relu(const float* __restrict__ AGG, const float* __restrict__ rdin,
                                   const float* __restrict__ bias, float* __restrict__ Hout) {
    const long t = (long)blockIdx.x * blockDim.x + threadIdx.x;
    if (t >= (long)NN * FEAT) return;
    const int i = (int)(t >> 7);
    const int f = (int)(t & 127);
    Hout[t] = fmaxf(AGG[t] * rdin[i] + bias[f], 0.0f);
}

// segment-sum pooling into emb region of d_out (emb is [B, 3*FEAT], coloff selects graph)
__global__ __launch_bounds__(256)
void pool_segsum(const float* __restrict__ H, const int* __restrict__ gid,
                 float* __restrict__ emb, int coloff) {
    const long t = (long)blockIdx.x * 256 + threadIdx.x;
    const int i = (int)(t >> 5);
    if (i >= NN) return;
    const int c = (int)(t & 31) << 2;
    const int g = gid[i];
    const float4 h4 = *(const float4*)(H + (size_t)i * FEAT + c);
    float* out = emb + (size_t)g * (3 * FEAT) + coloff + c;
    atomicAdd(out + 0, h4.x);
    atomicAdd(out + 1, h4.y);
    atomicAdd(out + 2, h4.z);
    atomicAdd(out + 3, h4.w);
}

// ---------------- head: BN (eval) + fc1 + relu ----------------
// one block per batch row; 128 threads (one per fc1 output)
__global__ __launch_bounds__(128)
void bn_fc1_relu(const float* __restrict__ emb, const float* __restrict__ gamma,
                 const float* __restrict__ beta, const float* __restrict__ mean,
                 const float* __restrict__ var, const float* __restrict__ fc1W,
                 const float* __restrict__ fc1b, float* __restrict__ hfc) {
    __shared__ float nb[3 * FEAT];
    const int b = blockIdx.x;
    for (int k = threadIdx.x; k < 3 * FEAT; k += 128) {
        nb[k] = (emb[b * (3 * FEAT) + k] - mean[k]) * rsqrtf(var[k] + EPSV) * gamma[k] + beta[k];
    }
    __syncthreads();
    const int j = threadIdx.x;
    float acc = fc1b[j];
    for (int k = 0; k < 3 * FEAT; ++k) acc += nb[k] * fc1W[k * FEAT + j];
    hfc[b * FEAT + j] = fmaxf(acc, 0.0f);
}

// fc2 + log_softmax; one thread per batch row
__global__ __launch_bounds__(256)
void fc2_logsoftmax(const float* __restrict__ hfc, const float* __restrict__ fc2W,
                    const float* __restrict__ fc2b, float* __restrict__ out) {
    const int b = threadIdx.x;   // single block of 256
    float lg[OUTC];
    for (int o = 0; o < OUTC; ++o) {
        float acc = fc2b[o];
        for (int k = 0; k < FEAT; ++k) acc += hfc[b * FEAT + k] * fc2W[k * OUTC + o];
        lg[o] = acc;
    }
    float mx = lg[0];
    for (int o = 1; o < OUTC; ++o) mx = fmaxf(mx, lg[o]);
    float se = 0.0f;
    for (int o = 0; o < OUTC; ++o) se += expf(lg[o] - mx);
    const float lse = logf(se) + mx;
    for (int o = 0; o < OUTC; ++o) out[b * OUTC + o] = lg[o] - lse;
}

// ---------------- launcher ----------------
extern "C" void kernel_launch(void* const* d_in, const int* in_sizes, int n_in,
                              void* d_out, int out_size, void* d_ws, size_t ws_size,
                              hipStream_t stream) {
    (void)in_sizes; (void)n_in; (void)out_size; (void)ws_size;

    // setup_inputs() dict order:
    const float* X[3]   = { (const float*)d_in[0], (const float*)d_in[4], (const float*)d_in[8]  };
    const int*   SRC[3] = { (const int*)d_in[1],   (const int*)d_in[5],   (const int*)d_in[9]   };
    const int*   DST[3] = { (const int*)d_in[2],   (const int*)d_in[6],   (const int*)d_in[10]  };
    const int*   GID[3] = { (const int*)d_in[3],   (const int*)d_in[7],   (const int*)d_in[11]  };
    const float* W1  = (const float*)d_in[12];
    const float* b1  = (const float*)d_in[13];
    const float* W2  = (const float*)d_in[14];
    const float* b2  = (const float*)d_in[15];
    const float* bng = (const float*)d_in[16];
    const float* bnb = (const float*)d_in[17];
    const float* bnm = (const float*)d_in[18];
    const float* bnv = (const float*)d_in[19];
    const float* f1W = (const float*)d_in[20];
    const float* f1b = (const float*)d_in[21];
    const float* f2W = (const float*)d_in[22];
    const float* f2b = (const float*)d_in[23];

    float* out = (float*)d_out;                 // [B,384] emb then [B,10] log-softmax
    float* logits_out = out + (size_t)BB * 3 * FEAT;

    // workspace layout (floats)
    float* ws   = (float*)d_ws;
    float* HW   = ws;                                   // [N,128] gemm output
    float* AGG  = HW  + (size_t)NN * FEAT;              // [N,128] aggregation
    float* H1   = AGG + (size_t)NN * FEAT;              // [N,128] activations
    float* rdo  = H1  + (size_t)NN * FEAT;              // [N] deg_out^-1/2
    float* rdi  = rdo + NN;                             // [N] deg_in^-1/2
    float* hfc  = rdi + NN;                             // [B,128]

    const int gemmBlocks = (NN / 16 + 7) / 8;           // 3125 tiles, 8 waves/block
    const int edgeBlocks = EE * 32 / 256;               // one wave per edge
    const int elemBlocks = (NN * FEAT) / 256;
    const int poolBlocks = NN * 32 / 256;

    // zero emb region of d_out (pool atomics accumulate into it)
    zero_f32<<<(BB * 3 * FEAT + 255) / 256, 256, 0, stream>>>(out, BB * 3 * FEAT);

    for (int g = 0; g < 3; ++g) {
        const int coloff = g * FEAT;
        // degrees (shared by both layers)
        zero_f32<<<(2 * NN + 255) / 256, 256, 0, stream>>>(rdo, 2 * NN);
        deg_count<<<(EE + 255) / 256, 256, 0, stream>>>(SRC[g], DST[g], rdo, rdi);
        deg_finalize<<<(NN + 255) / 256, 256, 0, stream>>>(rdo, rdi);

        // ---- layer 1: h1 = relu(conv(x, W1, b1)) ----
        gemm_wmma_f32<<<gemmBlocks, 256, 0, stream>>>(X[g], W1, HW, NN);
        zero_f32<<<elemBlocks, 256, 0, stream>>>(AGG, NN * FEAT);
        edge_scatter<<<edgeBlocks, 256, 0, stream>>>(HW, SRC[g], DST[g], rdo, AGG);
        conv_finalize_relu<<<elemBlocks, 256, 0, stream>>>(AGG, rdi, b1, H1);

        // ---- layer 2: h2 = relu(conv(h1, W2, b2)) ----
        gemm_wmma_f32<<<gemmBlocks, 256, 0, stream>>>(H1, W2, HW, NN);
        zero_f32<<<elemBlocks, 256, 0, stream>>>(AGG, NN * FEAT);
        edge_scatter<<<edgeBlocks, 256, 0, stream>>>(HW, SRC[g], DST[g], rdo, AGG);
        conv_finalize_relu<<<elemBlocks, 256, 0, stream>>>(AGG, rdi, b2, H1);

        // ---- sum pooling into emb columns [coloff, coloff+128) ----
        pool_segsum<<<poolBlocks, 256, 0, stream>>>(H1, GID[g], out, coloff);
    }

    // ---- head ----
    bn_fc1_relu<<<BB, 128, 0, stream>>>(out, bng, bnb, bnm, bnv, f1W, f1b, hfc);
    fc2_logsoftmax<<<1, BB, 0, stream>>>(hfc, f2W, f2b, logits_out);
}